// StegoScore_54958401519634
// MI455X (gfx1250) — compile-verified
//
#include <hip/hip_runtime.h>
#include <math.h>

typedef __attribute__((ext_vector_type(2))) float v2f;
typedef __attribute__((ext_vector_type(8))) float v8f;

#define H_IMG 512
#define PLANE_ELEMS (512 * 512)
#define NPLANES 24

// LDS tile geometry for a 32x32 rho tile:
//  xp patch : 62x62 valid (padded 64x64, pad rows/cols zeroed)
//  T1       : 47x62 valid (48x64)
//  |R|      : 47x47 valid (48x48)
//  T2       : 32x47 valid (32x48)
#define XP_S 64
#define T1_S 64
#define RB_S 48
#define T2_S 48

__device__ __forceinline__ int mirror_idx(int t) {
  t = (t < 0) ? -t : t;
  t = (t > 511) ? (1022 - t) : t;
  return t;
}

// Vertical 16-tap conv tile via WMMA: D[m,n] = sum_k band[m,k] * In[m0+k, n0+n]
// band is 16x32 (banded: band[m][k] = u[k-m] if 0<=k-m<16 else 0)
__device__ __forceinline__ v8f wmma_vert(const float* in, int stride, int m0, int n0,
                                         const float* aband, int lane) {
  const int m  = lane & 15;            // A row / B col for this lane
  const int kb = (lane >> 4) << 1;     // K sub-slot per f32 WMMA layout
  v8f acc = {0.f, 0.f, 0.f, 0.f, 0.f, 0.f, 0.f, 0.f};
#pragma unroll
  for (int k0 = 0; k0 < 32; k0 += 4) {
    v2f a, b;
    a.x = aband[m * 32 + (k0 + kb)];
    a.y = aband[m * 32 + (k0 + kb + 1)];
    b.x = in[(m0 + k0 + kb) * stride + n0 + m];
    b.y = in[(m0 + k0 + kb + 1) * stride + n0 + m];
    acc = __builtin_amdgcn_wmma_f32_16x16x4_f32(false, a, false, b, (short)0, acc,
                                                false, false);
  }
  return acc;
}

// Horizontal 16-tap conv tile via WMMA: D[m,n] = sum_k In[m0+m, n0+k] * band[k,n]
// band is 32x16 (banded: band[k][n] = v[k-n] if 0<=k-n<16 else 0)
__device__ __forceinline__ v8f wmma_horz(const float* in, int stride, int m0, int n0,
                                         const float* bband, int lane) {
  const int m  = lane & 15;
  const int kb = (lane >> 4) << 1;
  v8f acc = {0.f, 0.f, 0.f, 0.f, 0.f, 0.f, 0.f, 0.f};
#pragma unroll
  for (int k0 = 0; k0 < 32; k0 += 4) {
    v2f a, b;
    a.x = in[(m0 + m) * stride + n0 + (k0 + kb)];
    a.y = in[(m0 + m) * stride + n0 + (k0 + kb + 1)];
    b.x = bband[(k0 + kb) * 16 + m];
    b.y = bband[(k0 + kb + 1) * 16 + m];
    acc = __builtin_amdgcn_wmma_f32_16x16x4_f32(false, a, false, b, (short)0, acc,
                                                false, false);
  }
  return acc;
}

__device__ __forceinline__ void store_tile(float* out, int stride, int m0, int n0,
                                           v8f acc, int lane, bool do_abs) {
  const int n     = lane & 15;
  const int rbase = m0 + ((lane >> 4) << 3);
#pragma unroll
  for (int j = 0; j < 8; ++j) {
    float v = acc[j];
    if (do_abs) v = fabsf(v);
    out[(rbase + j) * stride + n0 + n] = v;
  }
}

__global__ __launch_bounds__(128)
void stego_conv_kernel(const float* __restrict__ x, const float* __restrict__ hpdf,
                       float* __restrict__ rho) {
  __shared__ float sXp[64 * XP_S];
  __shared__ float sT1[48 * T1_S];
  __shared__ float sRb[48 * RB_S];
  __shared__ float sT2[32 * T2_S];
  __shared__ float sAv[16 * 32];   // conv1 vertical band
  __shared__ float sBh[32 * 16];   // conv1 horizontal band
  __shared__ float sAv2[16 * 32];  // conv2 vertical band (|u| flipped)
  __shared__ float sBh2[32 * 16];  // conv2 horizontal band (|v| flipped)
  __shared__ float sH[16], sL[16], sU[16], sV[16], sGU[16], sGV[16];

  const int tid   = threadIdx.x;
  const int wave  = tid >> 5;
  const int lane  = tid & 31;
  const int plane = blockIdx.z;
  const int v0    = blockIdx.x * 32;
  const int u0    = blockIdx.y * 32;
  const float* xplane = x + (size_t)plane * PLANE_ELEMS;

  // 1D kernels: h = hpdf, l[i] = (-1)^i * h[15-i]
  if (tid < 16) {
    float h = hpdf[tid];
    sH[tid] = h;
    sL[tid] = ((tid & 1) ? -1.f : 1.f) * hpdf[15 - tid];
  }

  // Load mirrored 62x62 patch; zero pad rows/cols 62..63 (keeps zero-coef
  // band reads finite).
  for (int idx = tid; idx < 64 * XP_S; idx += 128) {
    int i = idx >> 6, j = idx & 63;
    float val = 0.f;
    if (i < 62 && j < 62) {
      int gr = mirror_idx(u0 - 15 + i);
      int gc = mirror_idx(v0 - 15 + j);
      val = xplane[gr * H_IMG + gc];
    }
    sXp[idx] = val;
  }

  float inv[8];
#pragma unroll
  for (int j = 0; j < 8; ++j) inv[j] = 0.f;

  // each wave owns one 16x16 quadrant of the 32x32 xi tile (for pass S4)
  const int sm0 = (wave >> 1) << 4;
  const int sn0 = (wave & 1) << 4;

  for (int f = 0; f < 3; ++f) {
    __syncthreads();  // sH/sL ready; previous filter's S4 done
    // filters: 0: LH = outer(l,h)  1: HL = outer(h,l)  2: HH = outer(h,h)
    if (tid < 16) {
      sU[tid]  = (f == 0) ? sL[tid] : sH[tid];
      sV[tid]  = (f == 1) ? sL[tid] : sH[tid];
      sGU[tid] = fabsf((f == 0) ? sL[15 - tid] : sH[15 - tid]);
      sGV[tid] = fabsf((f == 1) ? sL[15 - tid] : sH[15 - tid]);
    }
    __syncthreads();
    // build banded matrices (16x32 vertical A, 32x16 horizontal B)
    for (int idx = tid; idx < 512; idx += 128) {
      int m = idx >> 5, k = idx & 31, d = k - m;
      bool on = (d >= 0) && (d < 16);
      sAv[idx]  = on ? sU[d] : 0.f;
      sAv2[idx] = on ? sGU[d] : 0.f;
      int kk = idx >> 4, nn = idx & 15, d2 = kk - nn;
      bool on2 = (d2 >= 0) && (d2 < 16);
      sBh[idx]  = on2 ? sV[d2] : 0.f;
      sBh2[idx] = on2 ? sGV[d2] : 0.f;
    }
    __syncthreads();

    // S1: vertical conv1: T1(48x62ish) = band_u * Xp   (3x4 tiles of 16x16)
    for (int t = wave; t < 12; t += 4) {
      int m0 = (t >> 2) * 16, n0 = (t & 3) * 16;
      v8f acc = wmma_vert(sXp, XP_S, m0, n0, sAv, lane);
      store_tile(sT1, T1_S, m0, n0, acc, lane, false);
    }
    __syncthreads();

    // S2: horizontal conv1 + abs: |R|(48x48) = |T1 * band_v|   (3x3 tiles)
    for (int t = wave; t < 9; t += 4) {
      int m0 = (t / 3) * 16, n0 = (t % 3) * 16;
      v8f acc = wmma_horz(sT1, T1_S, m0, n0, sBh, lane);
      store_tile(sRb, RB_S, m0, n0, acc, lane, true);
    }
    __syncthreads();

    // S3: vertical conv2: T2(32x48) = band_gu * |R|   (2x3 tiles)
    for (int t = wave; t < 6; t += 4) {
      int m0 = (t / 3) * 16, n0 = (t % 3) * 16;
      v8f acc = wmma_vert(sRb, RB_S, m0, n0, sAv2, lane);
      store_tile(sT2, T2_S, m0, n0, acc, lane, false);
    }
    __syncthreads();

    // S4: horizontal conv2 -> xi quadrant (per wave), accumulate 1/xi
    v8f xi = wmma_horz(sT2, T2_S, sm0, sn0, sBh2, lane);
#pragma unroll
    for (int j = 0; j < 8; ++j) inv[j] += 1.0f / xi[j];
  }

  // store rho with roll(+1,+1) folded in; clamp/NaN -> 1e10
  float* rplane = rho + (size_t)plane * PLANE_ELEMS;
  const int n     = lane & 15;
  const int rbase = sm0 + ((lane >> 4) << 3);
#pragma unroll
  for (int j = 0; j < 8; ++j) {
    int uu = u0 + rbase + j;
    int vv = v0 + sn0 + n;
    float r = inv[j];
    if (!(r < 1e10f)) r = 1e10f;  // catches +inf, >wet, and NaN
    rplane[((uu + 1) & 511) * H_IMG + ((vv + 1) & 511)] = r;
  }
}

__global__ __launch_bounds__(256)
void stego_minmax_kernel(const float* __restrict__ rho, float* __restrict__ mn,
                         float* __restrict__ mx) {
  __shared__ float smn[256], smx[256];
  const int p = blockIdx.x;
  const float* base = rho + (size_t)p * PLANE_ELEMS;
  float lo = 3.402823466e38f, hi = -3.402823466e38f;
  for (int i = threadIdx.x; i < PLANE_ELEMS; i += 256) {
    float v = base[i];
    lo = fminf(lo, v);
    hi = fmaxf(hi, v);
  }
  smn[threadIdx.x] = lo;
  smx[threadIdx.x] = hi;
  __syncthreads();
  for (int s = 128; s > 0; s >>= 1) {
    if (threadIdx.x < s) {
      smn[threadIdx.x] = fminf(smn[threadIdx.x], smn[threadIdx.x + s]);
      smx[threadIdx.x] = fmaxf(smx[threadIdx.x], smx[threadIdx.x + s]);
    }
    __syncthreads();
  }
  if (threadIdx.x == 0) {
    mn[p] = smn[0];
    mx[p] = smx[0];
  }
}

__global__ __launch_bounds__(256)
void stego_final_kernel(const float* __restrict__ x, const float* __restrict__ rho,
                        const float* __restrict__ mn, const float* __restrict__ mx,
                        float* __restrict__ out, int ntot) {
  int i = blockIdx.x * blockDim.x + threadIdx.x;
  if (i < ntot) {
    int p = i >> 18;  // 512*512 = 2^18
    float lo = mn[p], hi = mx[p];
    float rn = (rho[i] - lo) / (hi - lo + 1e-8f);
    float z = 1.f - rn;
    float score = 1.f / (1.f + expf(-z));
    out[i] = x[i] * score;
  }
}

extern "C" void kernel_launch(void* const* d_in, const int* in_sizes, int n_in,
                              void* d_out, int out_size, void* d_ws, size_t ws_size,
                              hipStream_t stream) {
  const float* x    = (const float*)d_in[0];
  const float* hpdf = (const float*)d_in[1];
  float* out = (float*)d_out;

  float* rho = (float*)d_ws;                       // 24 * 512*512 floats
  float* mn  = rho + (size_t)NPLANES * PLANE_ELEMS;
  float* mx  = mn + NPLANES;

  dim3 g1(16, 16, NPLANES);
  stego_conv_kernel<<<g1, 128, 0, stream>>>(x, hpdf, rho);

  stego_minmax_kernel<<<NPLANES, 256, 0, stream>>>(rho, mn, mx);

  int ntot = NPLANES * PLANE_ELEMS;
  stego_final_kernel<<<(ntot + 255) / 256, 256, 0, stream>>>(x, rho, mn, mx, out, ntot);
}